// SCN_44942537786190
// MI455X (gfx1250) — compile-verified
//
#include <hip/hip_runtime.h>
#include <hip/hip_bf16.h>

typedef __attribute__((ext_vector_type(2))) float v2f;
typedef __attribute__((ext_vector_type(4))) float v4f;
typedef __attribute__((ext_vector_type(8))) float v8f;

#define B_N     4096
#define V_N     64
#define D_N     63
#define DEPTH_N 16

__global__ __launch_bounds__(256)
void scn_depthstep_kernel(const float* __restrict__ inp,     // (B,63)
                          const float* __restrict__ Lg,      // (16,64)
                          const float* __restrict__ vfs,     // (64,)
                          const float* __restrict__ vu,      // (64,63)
                          const float* __restrict__ biases,  // (16,)
                          float* __restrict__ out,           // (B,)
                          float* __restrict__ hold,          // (16,B,64,63)
                          float* __restrict__ hsel_out)      // (16,B,63)
{
    __shared__ __attribute__((aligned(16))) float h_lds[V_N * D_N];     // 4032 fl, contiguous (v,d)
    __shared__ float L_lds[DEPTH_N * V_N];                              // 1024 fl
    __shared__ float w_lds[V_N];
    __shared__ float f_lds[V_N];
    __shared__ float hsel_lds[V_N];
    __shared__ int   idx_lds;

    const int tid  = threadIdx.x;
    const int lane = tid & 31;
    const int wave = tid >> 5;
    const int b    = blockIdx.x;

    // ---- init: L and h=visible_units into LDS, f=visible_fs, w=cat(1-sum(inp),inp)
    for (int j = tid; j < DEPTH_N * V_N; j += 256) L_lds[j] = Lg[j];
    for (int j = tid; j < V_N * D_N;    j += 256) h_lds[j] = vu[j];
    if (tid < V_N) f_lds[tid] = vfs[tid];
    if (wave == 4) {
        const float* row = inp + (size_t)b * D_N;
        float e = 0.0f;
        if (lane < D_N)      e += row[lane];
        if (lane + 32 < D_N) e += row[lane + 32];
        for (int off = 16; off > 0; off >>= 1) e += __shfl_xor(e, off, 32);
        if (lane == 0)       w_lds[0]         = 1.0f - e;
        if (lane < D_N)      w_lds[lane + 1]  = row[lane];
        if (lane + 32 < D_N) w_lds[lane + 33] = row[lane + 32];
    }
    __syncthreads();

    // ---- preload A = L (16 depth-rows x 64 K) for WMMA f32 16x16x4:
    // A chunk k: lane m(=lane&15) holds L[m][4k+kb], L[m][4k+kb+1]; kb=0 (lanes 0-15) / 2 (16-31)
    const int m  = lane & 15;
    const int kb = (lane < 16) ? 0 : 2;
    v2f a_reg[16];
    if (wave < 4) {
        #pragma unroll
        for (int k = 0; k < 16; ++k) {
            a_reg[k].x = L_lds[m * V_N + 4 * k + kb];
            a_reg[k].y = L_lds[m * V_N + 4 * k + kb + 1];
        }
    }
    // B-tile column this lane owns; col 63 is discarded padding -> clamp address
    // (it only ever feeds output column 63, which we never read), so all B
    // loads are unconditional: no exec-mask juggling in the hot loop.
    const int colr = (wave << 4) + m;                 // 0..63
    const bool cok = (colr < D_N);
    const int col  = cok ? colr : 0;

    for (int i = 0; i < DEPTH_N; ++i) {
        // ===== phase A: everyone reads pre-update state =====
        // (1) stream h_old[i][b] : 4032 contiguous floats, NT b128 stores
        {
            float* dst = hold + ((size_t)i * B_N + b) * (size_t)(V_N * D_N);
            const v4f* src = (const v4f*)h_lds;
            for (int t = tid; t < (V_N * D_N) / 4; t += 256) {
                v4f val = src[t];
                __builtin_nontemporal_store(val, ((v4f*)dst) + t);
            }
        }
        if (wave < 4) {
            // (2) P = L(16x64) @ h(64x64pad); this wave owns N-tile 'wave'.
            // Prefetch all 16 B chunks, then run the chained WMMAs.
            v2f breg[16];
            #pragma unroll
            for (int k = 0; k < 16; ++k) {
                const int r0 = 4 * k + kb;
                breg[k].x = h_lds[r0 * D_N + col];
                breg[k].y = h_lds[(r0 + 1) * D_N + col];
            }
            v8f acc = {0.f, 0.f, 0.f, 0.f, 0.f, 0.f, 0.f, 0.f};
            #pragma unroll
            for (int k = 0; k < 16; ++k) {
                acc = __builtin_amdgcn_wmma_f32_16x16x4_f32(
                          false, a_reg[k], false, breg[k], (short)0, acc, false, false);
            }
            // extract row i -> hsel_lds
            float rv = acc[0];
            #pragma unroll
            for (int j = 1; j < 8; ++j) if ((i & 7) == j) rv = acc[j];
            const bool mine = (i < 8) ? (lane < 16) : (lane >= 16);
            if (mine && cok) hsel_lds[colr] = rv;
        } else if (wave == 4) {
            // (3) argmin(wd) with first-index ties, w/f updates
            const float Lv0 = L_lds[i * V_N + lane];
            const float Lv1 = L_lds[i * V_N + lane + 32];
            const float w0v = w_lds[lane],  w1v = w_lds[lane + 32];
            const float f0v = f_lds[lane],  f1v = f_lds[lane + 32];
            const float wd0 = w0v / (Lv0 + 1e-20f);
            const float wd1 = w1v / (Lv1 + 1e-20f);
            float mv = (wd0 <= wd1) ? wd0  : wd1;
            int   mi = (wd0 <= wd1) ? lane : lane + 32;
            for (int off = 16; off > 0; off >>= 1) {
                float ov = __shfl_xor(mv, off, 32);
                int   oi = __shfl_xor(mi, off, 32);
                if (ov < mv || (ov == mv && oi < mi)) { mv = ov; mi = oi; }
            }
            float t = Lv0 * f0v + Lv1 * f1v;
            for (int off = 16; off > 0; off >>= 1) t += __shfl_xor(t, off, 32);
            const float fsel = t + biases[i];
            w_lds[lane]      = (lane == mi)      ? mv : (w0v - mv * Lv0);
            w_lds[lane + 32] = (lane + 32 == mi) ? mv : (w1v - mv * Lv1);
            if (lane == 0) { f_lds[mi] = fsel; idx_lds = mi; }
        }
        __syncthreads();
        // ===== phase B: commit h row update + stream h_sel =====
        if (tid < D_N) {
            const float hv = hsel_lds[tid];
            h_lds[idx_lds * D_N + tid] = hv;
            __builtin_nontemporal_store(
                hv, hsel_out + ((size_t)i * B_N + b) * (size_t)D_N + tid);
        }
        __syncthreads();
    }

    // ---- out[b] = dot(w, f)
    if (wave == 4) {
        float t = w_lds[lane] * f_lds[lane] + w_lds[lane + 32] * f_lds[lane + 32];
        for (int off = 16; off > 0; off >>= 1) t += __shfl_xor(t, off, 32);
        if (lane == 0) out[b] = t;
    }
}

extern "C" void kernel_launch(void* const* d_in, const int* in_sizes, int n_in,
                              void* d_out, int out_size, void* d_ws, size_t ws_size,
                              hipStream_t stream) {
    const float* inp    = (const float*)d_in[0];
    const float* L      = (const float*)d_in[1];
    const float* vfs    = (const float*)d_in[2];
    const float* vu     = (const float*)d_in[3];
    const float* biases = (const float*)d_in[4];

    float* out  = (float*)d_out;
    float* hold = out + B_N;                                    // after 4096 floats of 'out'
    float* hsel = hold + (size_t)DEPTH_N * B_N * V_N * D_N;     // after 264,241,152 floats of h_old

    scn_depthstep_kernel<<<B_N, 256, 0, stream>>>(inp, L, vfs, vu, biases,
                                                  out, hold, hsel);
}